// YOLOV9Loss_76287209111907
// MI455X (gfx1250) — compile-verified
//
#include <hip/hip_runtime.h>

typedef float v2f __attribute__((ext_vector_type(2)));
typedef float v8f __attribute__((ext_vector_type(8)));

#define F_EPS 1e-9f
#define NB 32
#define NA 8400
#define NC 80
#define NR 16
#define BA (NB * NA)

__device__ __forceinline__ float bce1(float x, float t) {
    return fmaxf(x, 0.f) - x * t + __logf(1.f + __expf(-fabsf(x)));
}

__device__ __forceinline__ float wave_sum(float v) {
#pragma unroll
    for (int off = 16; off > 0; off >>= 1) v += __shfl_xor(v, off, 32);
    return v;
}

__global__ void yolo_zero_acc(float* acc) {
    if (threadIdx.x < 4) acc[threadIdx.x] = 0.f;
}

__global__ __launch_bounds__(256) void yolo_loss_main(
    const float* __restrict__ pcls, const float* __restrict__ panc,
    const float* __restrict__ pboxg, const float* __restrict__ tgt,
    const int* __restrict__ vmask, const float* __restrict__ agrid,
    const float* __restrict__ scaler, float* __restrict__ acc) {

    const int idx = blockIdx.x * blockDim.x + threadIdx.x;
    float s_bce = 0.f, s_t = 0.f, s_iou = 0.f, s_dfl = 0.f;

    if (idx < BA) {
        const int a = idx % NA;
        const float inv = 1.0f / scaler[a];

        const float4* cp4 = (const float4*)(pcls + (size_t)idx * NC);
        const float4* tp4 = (const float4*)(tgt + (size_t)idx * (NC + 4));
        const float4* ap4 = (const float4*)(panc + (size_t)idx * (4 * NR));
        __builtin_prefetch(panc + (size_t)idx * (4 * NR), 0, 0);  // global_prefetch_b8

        // ---- BCE over 80 classes + targets_cls row-sum ----
#pragma unroll 5
        for (int i = 0; i < NC / 4; ++i) {
            float4 x = cp4[i];
            float4 t = tp4[i];
            s_bce += bce1(x.x, t.x) + bce1(x.y, t.y) + bce1(x.z, t.z) + bce1(x.w, t.w);
            s_t += t.x + t.y + t.z + t.w;
        }

        // ---- CIoU (boxes normalized by scaler) ----
        float4 pb = ((const float4*)pboxg)[idx];
        float4 tb = tp4[NC / 4];  // targets box = floats 80..83 of the 84-float row
        float p1 = pb.x * inv, p2 = pb.y * inv, p3 = pb.z * inv, p4 = pb.w * inv;
        float t1 = tb.x * inv, t2 = tb.y * inv, t3 = tb.z * inv, t4 = tb.w * inv;

        float iw = fmaxf(fminf(p3, t3) - fmaxf(p1, t1), 0.f);
        float ih = fmaxf(fminf(p4, t4) - fmaxf(p2, t2), 0.f);
        float inter = iw * ih;
        float a1 = (p3 - p1) * (p4 - p2);
        float a2 = (t3 - t1) * (t4 - t2);
        float uni = a1 + a2 - inter + F_EPS;
        float iou = inter / uni;
        float dcx = 0.5f * ((p1 + p3) - (t1 + t3));
        float dcy = 0.5f * ((p2 + p4) - (t2 + t4));
        float cent = dcx * dcx + dcy * dcy;
        float ew = fmaxf(p3, t3) - fminf(p1, t1);
        float eh = fmaxf(p4, t4) - fminf(p2, t2);
        float diag = ew * ew + eh * eh + F_EPS;
        float diou = iou - cent / diag;
        float ar = atanf((p3 - p1) / (p4 - p2 + F_EPS)) - atanf((t3 - t1) / (t4 - t2 + F_EPS));
        float vv = (4.0f / (float)(M_PI * M_PI)) * ar * ar;
        float alpha = vv / (vv - iou + 1.0f + F_EPS);
        float ciou = diou - alpha * vv;

        float msk = (vmask[idx] != 0) ? 1.f : 0.f;
        float box_norm = s_t * msk;
        s_iou = (1.0f - ciou) * box_norm;

        // ---- DFL: 4 sides, log-softmax over 16 bins ----
        float gx = agrid[2 * a + 0] * inv;
        float gy = agrid[2 * a + 1] * inv;
        float tdist[4] = {gx - t1, gy - t2, t3 - gx, t4 - gy};
        float dfl = 0.f;
#pragma unroll
        for (int sd = 0; sd < 4; ++sd) {
            float td = fminf(fmaxf(tdist[sd], 0.f), (float)NR - 1.01f);
            int li = (int)floorf(td);
            float fl = (float)li;
            float4 q0 = ap4[sd * 4 + 0], q1 = ap4[sd * 4 + 1];
            float4 q2 = ap4[sd * 4 + 2], q3 = ap4[sd * 4 + 3];
            float v[16] = {q0.x, q0.y, q0.z, q0.w, q1.x, q1.y, q1.z, q1.w,
                           q2.x, q2.y, q2.z, q2.w, q3.x, q3.y, q3.z, q3.w};
            float mx = v[0];
#pragma unroll
            for (int k = 1; k < 16; ++k) mx = fmaxf(mx, v[k]);
            float sm = 0.f, vl = 0.f, vr = 0.f;
#pragma unroll
            for (int k = 0; k < 16; ++k) {
                sm += __expf(v[k] - mx);
                vl = (k == li) ? v[k] : vl;
                vr = (k == li + 1) ? v[k] : vr;
            }
            float lse = mx + __logf(sm);
            dfl += (lse - vl) * (fl + 1.f - td) + (lse - vr) * (td - fl);
        }
        s_dfl = 0.25f * dfl * box_norm;
    }

    // ---- hierarchical reduction: lanes -> waves -> WMMA across waves ----
    float p0 = wave_sum(s_bce);
    float p1s = wave_sum(s_t);
    float p2 = wave_sum(s_iou);
    float p3s = wave_sum(s_dfl);

    __shared__ float wpart[32];  // 8 waves x 4 quantities
    const int lane = threadIdx.x & 31;
    const int wid = threadIdx.x >> 5;
    if (lane == 0) {
        wpart[wid * 4 + 0] = p0;
        wpart[wid * 4 + 1] = p1s;
        wpart[wid * 4 + 2] = p2;
        wpart[wid * 4 + 3] = p3s;
    }
    __syncthreads();

    if (wid == 0) {
        // A(16x4) f32 layout: lanes 0-15 -> K=0 (VGPR0), K=1 (VGPR1);
        // lanes 16-31 -> K=2 (VGPR0), K=3 (VGPR1).  Set A[m,k] = wpart[wave=k][q=m].
        const int h = lane >> 4;     // 0 or 1 -> K pair
        const int q = lane & 15;     // row m
        const bool ok = (q < 4);
        v2f A1, A2, Bones;
        A1.x = ok ? wpart[(2 * h + 0) * 4 + q] : 0.f;  // waves 0 / 2
        A1.y = ok ? wpart[(2 * h + 1) * 4 + q] : 0.f;  // waves 1 / 3
        A2.x = ok ? wpart[(2 * h + 4) * 4 + q] : 0.f;  // waves 4 / 6
        A2.y = ok ? wpart[(2 * h + 5) * 4 + q] : 0.f;  // waves 5 / 7
        Bones.x = 1.f;
        Bones.y = 1.f;
        v8f C = {};
        // D[m,n] = sum_k A[m,k] : exact f32 row-sum of wave partials
        C = __builtin_amdgcn_wmma_f32_16x16x4_f32(false, A1, false, Bones,
                                                  (short)0, C, false, false);
        C = __builtin_amdgcn_wmma_f32_16x16x4_f32(false, A2, false, Bones,
                                                  (short)0, C, false, false);
        // lane 0 holds D[m=q, n=0] in C[q] for q = 0..3
        if (lane == 0) {
            atomicAdd(&acc[0], C[0]);  // bce sum
            atomicAdd(&acc[1], C[1]);  // cls_norm
            atomicAdd(&acc[2], C[2]);  // (1-ciou)*box_norm sum
            atomicAdd(&acc[3], C[3]);  // dfl*box_norm sum
        }
    }
}

__global__ void yolo_finalize(const float* __restrict__ acc, float* __restrict__ out) {
    if (threadIdx.x == 0) {
        float cn = acc[1];
        out[0] = acc[2] / cn;  // loss_iou
        out[1] = acc[3] / cn;  // loss_dfl
        out[2] = acc[0] / cn;  // loss_cls
    }
}

extern "C" void kernel_launch(void* const* d_in, const int* in_sizes, int n_in,
                              void* d_out, int out_size, void* d_ws, size_t ws_size,
                              hipStream_t stream) {
    const float* pcls  = (const float*)d_in[0];  // (32,8400,80)
    const float* panc  = (const float*)d_in[1];  // (32,8400,4,16)
    const float* pbox  = (const float*)d_in[2];  // (32,8400,4)
    const float* tgt   = (const float*)d_in[3];  // (32,8400,84)
    const int*   vmask = (const int*)d_in[4];    // (32,8400) bool->int
    const float* agrid = (const float*)d_in[5];  // (8400,2)
    const float* scal  = (const float*)d_in[6];  // (8400,)
    float* out = (float*)d_out;
    float* acc = (float*)d_ws;  // 4 floats of scratch accumulators

    yolo_zero_acc<<<1, 64, 0, stream>>>(acc);
    const int threads = 256;
    const int blocks = (BA + threads - 1) / threads;  // 1050
    yolo_loss_main<<<blocks, threads, 0, stream>>>(pcls, panc, pbox, tgt, vmask,
                                                   agrid, scal, acc);
    yolo_finalize<<<1, 32, 0, stream>>>(acc, out);
}